// PINN_LORA_55087250539187
// MI455X (gfx1250) — compile-verified
//
#include <hip/hip_runtime.h>

// CDNA5 / gfx1250 TensoRF line-decomposition kernel.
// p[n,r] = prod_a lerp(line_vec[a][r], coord);  out = p @ W^T  via V_WMMA_F32_16X16X4_F32.

typedef __attribute__((ext_vector_type(2))) float v2f;
typedef __attribute__((ext_vector_type(8))) float v8f;

#define RANK  16
#define GRIDW 128
#define NOUT  3

__global__ __launch_bounds__(256) void PINN_LORA_55087250539187_kernel(
    const float* __restrict__ x,        // [N,3]
    const float* __restrict__ line_vec, // [3,16,128]
    const float* __restrict__ W,        // [3,16]
    float* __restrict__ out,            // [N,3]
    int nTiles)                          // N/16 tiles of 16 points
{
    // LDS: table stored as adjacent pairs so each lerp tap is ONE aligned ds_load_b64.
    // s_pair[(a*16+r)*128 + g] = { line[a][r][g], line[a][r][min(g+1,127)] }  -> 48 KB
    __shared__ v2f s_pair[3 * RANK * GRIDW];

    const int tid = threadIdx.x;
    for (int i = tid; i < 3 * RANK * GRIDW; i += blockDim.x) {
        int g = i & (GRIDW - 1);
        float v0 = line_vec[i];
        float v1 = line_vec[(g < GRIDW - 1) ? (i + 1) : i];
        v2f pr; pr.x = v0; pr.y = v1;
        s_pair[i] = pr;
    }
    __syncthreads();

    const int lane = tid & 31;          // wave32
    const int n    = lane & 15;         // point-in-tile (A rows) == output column (D cols)
    const int kh   = (lane >> 4) << 1;  // K sub-offset: lanes 0-15 -> K{0,1}, lanes 16-31 -> K{2,3}

    // B fragments: chunk j is W^T rows 4j..4j+3 (K) x 16 cols (N); cols >= 3 are zero.
    // f32 4x16 B layout mirrors A: lane holds {K=4j+kh, K=4j+kh+1} at column n.
    v2f bmat[4];
#pragma unroll
    for (int j = 0; j < 4; ++j) {
        float b0 = 0.0f, b1 = 0.0f;
        if (n < NOUT) {
            b0 = W[n * RANK + 4 * j + kh + 0];
            b1 = W[n * RANK + 4 * j + kh + 1];
        }
        bmat[j].x = b0; bmat[j].y = b1;
    }

    const int wavesPerBlock = blockDim.x >> 5;
    const int waveId     = blockIdx.x * wavesPerBlock + (tid >> 5);
    const int waveStride = gridDim.x * wavesPerBlock;

    for (int tile = waveId; tile < nTiles; tile += waveStride) {
        // Each lane loads its point's coords (lanes L and L+16 share a point; dup
        // requests to the same cacheline merge, and x fits entirely in the 192MB L2).
        const long p = (long)tile * 16 + n;
        const float* xp = x + p * 3;
        float coord[3];
        coord[0] = xp[2];   // vecMode: line axis 0 <- x[:,2]
        coord[1] = xp[1];
        coord[2] = xp[0];

        // prod[j] = A-fragment chunk j: {rank 4j+kh, rank 4j+kh+1} of point n.
        v2f prod[4];
#pragma unroll
        for (int j = 0; j < 4; ++j) { prod[j].x = 1.0f; prod[j].y = 1.0f; }

#pragma unroll
        for (int a = 0; a < 3; ++a) {
            float pos = (coord[a] + 1.0f) * (0.5f * (GRIDW - 1));
            float f   = floorf(pos);
            f = fminf(fmaxf(f, 0.0f), (float)(GRIDW - 1));
            int   i0  = (int)f;
            float w   = pos - f;
            int base  = (a * RANK + kh) * GRIDW + i0;
#pragma unroll
            for (int j = 0; j < 4; ++j) {
#pragma unroll
                for (int t = 0; t < 2; ++t) {
                    v2f v = s_pair[base + (4 * j + t) * GRIDW]; // ds_load_b64, aligned
                    float val = v.x + w * (v.y - v.x);
                    prod[j][t] *= val;
                }
            }
        }

        // D[16 pts x 16 ch] = sum_j A_j(16x4) x B_j(4x16); 4 chained f32 WMMAs.
        v8f acc = {};
        acc = __builtin_amdgcn_wmma_f32_16x16x4_f32(false, prod[0], false, bmat[0], (short)0, acc, false, false);
        acc = __builtin_amdgcn_wmma_f32_16x16x4_f32(false, prod[1], false, bmat[1], (short)0, acc, false, false);
        acc = __builtin_amdgcn_wmma_f32_16x16x4_f32(false, prod[2], false, bmat[2], (short)0, acc, false, false);
        acc = __builtin_amdgcn_wmma_f32_16x16x4_f32(false, prod[3], false, bmat[3], (short)0, acc, false, false);

        // D layout: VGPR v, lane L -> point (v + 8*(L>>4)), channel (L&15).
        if (n < NOUT) {
            const long ob = (long)tile * 16 + ((lane >> 4) << 3);
#pragma unroll
            for (int v = 0; v < 8; ++v) {
                out[(ob + v) * 3 + n] = acc[v];
            }
        }
    }
}

extern "C" void kernel_launch(void* const* d_in, const int* in_sizes, int n_in,
                              void* d_out, int out_size, void* d_ws, size_t ws_size,
                              hipStream_t stream) {
    (void)n_in; (void)out_size; (void)d_ws; (void)ws_size;
    const float* x        = (const float*)d_in[0];  // [N,3]
    const float* line_vec = (const float*)d_in[1];  // [3,16,128]
    const float* W        = (const float*)d_in[2];  // [3,16]
    float* out = (float*)d_out;                     // [N,3]

    const int nPoints = in_sizes[0] / 3;            // 4,194,304
    const int nTiles  = nPoints >> 4;               // divisible by 16

    // 2048 blocks x 8 wave32s = 16384 waves; 16 tiles per wave, grid-stride.
    PINN_LORA_55087250539187_kernel<<<2048, 256, 0, stream>>>(x, line_vec, W, out, nTiles);
}